// FM_FTRL_4904852652524
// MI455X (gfx1250) — compile-verified
//
#include <hip/hip_runtime.h>

#define ETA    1e-5f
#define D      1024
#define DA     1023        // a = x[:-1]
#define M2     512         // two_m
#define NSTEPS 20000
#define BLK    32          // time-block size
#define NBLK   (NSTEPS / BLK)   // 625
#define GW2_LD 1024        // g_W2 padded leading dim; column 1023 is always 0
#define NWG    48          // persistent grid size (co-resident)

typedef float v2f __attribute__((ext_vector_type(2)));
typedef float v8f __attribute__((ext_vector_type(8)));

#ifndef __has_builtin
#define __has_builtin(x) 0
#endif
#if __has_builtin(__builtin_amdgcn_tensor_load_to_lds) && \
    __has_builtin(__builtin_amdgcn_s_wait_tensorcnt)
#define USE_TDM 1
#else
#define USE_TDM 0
#endif

__device__ __forceinline__ v8f wmma_f32_4(v2f a, v2f b, v8f c) {
  // V_WMMA_F32_16X16X4_F32 : D(16x16,f32) = A(16x4,f32) * B(4x16,f32) + C
  return __builtin_amdgcn_wmma_f32_16x16x4_f32(false, a, false, b, (short)0, c,
                                               false, false);
}

#if USE_TDM
typedef unsigned int u32x4 __attribute__((ext_vector_type(4)));
typedef int i32x4 __attribute__((ext_vector_type(4)));
typedef int i32x8 __attribute__((ext_vector_type(8)));

// TDM: load 2D tile (BLK x D, 4B elems, row stride NSTEPS) from global to LDS.
// D# layout per CDNA5 ISA 8.3/8.4: group0 {count,lds_addr,global_addr,type=2},
// group1 {data_size=4B, tensor_dim0/1, tile_dim0/1, tensor_dim0_stride}.
__device__ __forceinline__ void tdm_load_Ablk(const float* gsrc, float* lds_dst) {
  const unsigned long long ga = (unsigned long long)(const void*)gsrc;
  u32x4 g0;
  g0[0] = 1u;                                           // count=1, user D#
  g0[1] = (unsigned int)(unsigned long long)(void*)lds_dst;  // LDS byte addr
  g0[2] = (unsigned int)ga;                             // global_addr[31:0]
  g0[3] = (unsigned int)((ga >> 32) & 0x1FFFFFFull) | (2u << 30); // [56:32]|type=2
  i32x8 g1;
  g1[0] = (int)(2u << 16);                              // data_size = 4B
  g1[1] = (int)(((unsigned)NSTEPS & 0xFFFFu) << 16);    // tensor_dim0 lo16
  g1[2] = (int)((((unsigned)NSTEPS >> 16) & 0xFFFFu) |
                (((unsigned)D & 0xFFFFu) << 16));       // dim0 hi16 | dim1 lo16
  g1[3] = (int)((((unsigned)D >> 16) & 0xFFFFu) |
                ((unsigned)BLK << 16));                 // dim1 hi16 | tile_dim0
  g1[4] = (int)((unsigned)D);                           // tile_dim1 ; tile_dim2=0
  g1[5] = (int)((unsigned)NSTEPS);                      // tensor_dim0_stride lo32
  g1[6] = 0;                                            // stride0 hi ; stride1 lo
  g1[7] = 0;
  const i32x4 gz = {0, 0, 0, 0};
#if defined(__clang_major__) && (__clang_major__ >= 23)
  const i32x8 gz8 = {0, 0, 0, 0, 0, 0, 0, 0};
  __builtin_amdgcn_tensor_load_to_lds(g0, g1, gz, gz, gz8, 0);
#else
  __builtin_amdgcn_tensor_load_to_lds(g0, g1, gz, gz, 0);
#endif
}
#endif

// Grid-wide generation barrier (all NWG workgroups co-resident).
__device__ __forceinline__ void grid_sync(int* cnt, int* gen) {
  __threadfence();
  __syncthreads();
  if (threadIdx.x == 0) {
    const int g = atomicAdd(gen, 0);
    if (atomicAdd(cnt, 1) == NWG - 1) {
      atomicExch(cnt, 0);
      __threadfence();
      atomicAdd(gen, 1);
    } else {
      while (atomicAdd(gen, 0) == g) __builtin_amdgcn_s_sleep(2);
    }
  }
  __syncthreads();
  __threadfence();
}

// ---------------------------------------------------------------------------
// Zero FTRL state + barrier words (must run every call: graph replays).
__global__ void k_init(float* __restrict__ gW2, float* __restrict__ gw1,
                       int* __restrict__ cnt, int* __restrict__ gen) {
  int idx = blockIdx.x * blockDim.x + threadIdx.x;
  int stride = gridDim.x * blockDim.x;
  for (int i = idx; i < M2 * GW2_LD; i += stride) gW2[i] = 0.f;
  if (idx < D) gw1[idx] = 0.f;
  if (idx == 0) { *cnt = 0; *gen = 0; }
}

// ---------------------------------------------------------------------------
// Step 0 is special: t0 = W2_init @ a_0, scalar0 = w1_init . x_0 + ||t0||^2
__global__ void k_step0(const float* __restrict__ At,
                        const float* __restrict__ w1i,
                        const float* __restrict__ W2i,
                        float* __restrict__ t0, float* __restrict__ s0) {
  __shared__ float red[16];
  int tid = threadIdx.x;                       // 512 threads; one row of W2 each
  const float* wrow = W2i + (size_t)tid * DA;
  float acc = 0.f;
  for (int r = 0; r < DA; ++r) acc += wrow[r] * At[(size_t)r * NSTEPS];
  t0[tid] = acc;
  float tt = acc * acc;
  for (int o = 16; o > 0; o >>= 1) tt += __shfl_xor(tt, o, 32);
  if ((tid & 31) == 0) red[tid >> 5] = tt;
  __syncthreads();
  if (tid < 32) {
    float v = (tid < 16) ? red[tid] : 0.f;
    for (int o = 16; o > 0; o >>= 1) v += __shfl_xor(v, o, 32);
    float dp = 0.f;
    for (int r = tid; r < D; r += 32) dp += w1i[r] * At[(size_t)r * NSTEPS];
    for (int o = 16; o > 0; o >>= 1) dp += __shfl_xor(dp, o, 32);
    if (tid == 0) s0[0] = dp + v;
  }
}

// ---------------------------------------------------------------------------
__global__ void k_bcopy(const float* __restrict__ b, float* __restrict__ out) {
  int i = blockIdx.x * blockDim.x + threadIdx.x;
  if (i < NSTEPS) out[NSTEPS + i] = b[i];
}

// ---------------------------------------------------------------------------
// Persistent fused scan: all NBLK block-iterations inside one kernel.
//   stage  : A_blk (1024x32, 128KB) -> LDS via TDM (or float4 fallback)
//   phase A: T_base = g_W2 @ A_blk (WMMA); Ga = A_blk^T A_blk (WMMA, masked);
//            u_base/xlast dots
//   phase B: 32-step sequential recurrence (workgroup 0 only)
//   phase C: g_W2 += 2*T @ A_blk^T (WMMA, all waves) ; g_w1 update
__global__ void __launch_bounds__(256)
k_ftrl_main(const float* __restrict__ At, const float* __restrict__ b,
            float* __restrict__ gW2, float* __restrict__ gw1,
            float* __restrict__ Tbase, float* __restrict__ Ga,
            float* __restrict__ ubase, float* __restrict__ xlast,
            float* __restrict__ Tout, float* __restrict__ signs,
            const float* __restrict__ t0, const float* __restrict__ s0,
            float* __restrict__ preds, int* cnt, int* gen) {
  __shared__ float Ash[D * BLK];                 // 128 KB: A_blk[r][j]
  __shared__ float ga[BLK * BLK];                // phase B (WG 0)
  __shared__ float xl[BLK], ub[BLK], sg[BLK], red[8];

  const int g    = blockIdx.x;
  const int tid  = threadIdx.x;
  const int wave = tid >> 5;
  const int lane = tid & 31;
  const int half = lane >> 4;
  const int mrow = lane & 15;
  const int gw   = g * 8 + wave;                 // global wave id, 0..383

  for (int blk = 0; blk < NBLK; ++blk) {
    const int s = blk * BLK;

    // ---- stage A_blk into LDS ----
#if USE_TDM
    if (wave == 0) {
      tdm_load_Ablk(At + s, Ash);
      __builtin_amdgcn_s_wait_tensorcnt(0);
    }
    __syncthreads();
#else
    for (int idx4 = tid; idx4 < (D * BLK) / 4; idx4 += 256) {
      const int r = idx4 >> 3, c4 = (idx4 & 7) << 2;   // 128B-contig rows
      *(float4*)&Ash[(r << 5) + c4] =
          *(const float4*)&At[(size_t)r * NSTEPS + s + c4];
    }
    __syncthreads();
#endif

    // ---- phase A ----
    if (gw < 64) {                               // T_base tiles
      const int rt = gw >> 1, ct = gw & 1;
      v8f c = {0.f, 0.f, 0.f, 0.f, 0.f, 0.f, 0.f, 0.f};
      const float* arow = gW2 + (size_t)(rt * 16 + mrow) * GW2_LD + 2 * half;
      const float* bsh  = Ash + ct * 16 + mrow;
      for (int k = 0; k < GW2_LD; k += 4) {
        v2f a  = {arow[k], arow[k + 1]};
        v2f bb = {bsh[(k + 2 * half) * BLK], bsh[(k + 2 * half + 1) * BLK]};
        c = wmma_f32_4(a, bb, c);
      }
      for (int rr = 0; rr < 8; ++rr)
        Tbase[(rt * 16 + rr + 8 * half) * BLK + ct * 16 + mrow] = c[rr];
    } else if (gw < 68) {                        // Gram tiles (mask row 1023)
      const int q = gw - 64, rt = q >> 1, ct = q & 1;
      v8f c = {0.f, 0.f, 0.f, 0.f, 0.f, 0.f, 0.f, 0.f};
      for (int k = 0; k < 1024; k += 4) {
        const int r0 = k + 2 * half;
        // loads unconditional (row 1023 valid LDS); mask as select, no branch
        float a0 = Ash[r0 * BLK + rt * 16 + mrow];
        float a1 = Ash[(r0 + 1) * BLK + rt * 16 + mrow];
        float b0 = Ash[r0 * BLK + ct * 16 + mrow];
        float b1 = Ash[(r0 + 1) * BLK + ct * 16 + mrow];
        if (r0 + 1 == DA) { a1 = 0.f; b1 = 0.f; }
        v2f a = {a0, a1}, bb = {b0, b1};
        c = wmma_f32_4(a, bb, c);
      }
      for (int rr = 0; rr < 8; ++rr)
        Ga[(rt * 16 + rr + 8 * half) * BLK + ct * 16 + mrow] = c[rr];
    } else if (gw < 76) {                        // u_base / xlast
      const int jw = gw - 68;
      for (int j = jw; j < BLK; j += 8) {
        float dp = 0.f;
        for (int r = lane; r < D; r += 32) dp += gw1[r] * Ash[r * BLK + j];
        for (int o = 16; o > 0; o >>= 1) dp += __shfl_xor(dp, o, 32);
        if (lane == 0) { ubase[j] = dp; xlast[j] = Ash[DA * BLK + j]; }
      }
    }
    grid_sync(cnt, gen);

    // ---- phase B: sequential 32-step recurrence (WG 0 only) ----
    if (g == 0) {
      for (int i = tid; i < BLK * BLK; i += 256) ga[i] = Ga[i];
      if (tid < BLK) { xl[tid] = xlast[tid]; ub[tid] = ubase[tid]; }
      __syncthreads();
      float tl0[BLK], tl1[BLK];                  // thread owns components p, p+256
      const int p0 = tid, p1 = tid + 256;
      const bool first = (blk == 0);
      for (int j = 0; j < BLK; ++j) {
        float a0 = Tbase[p0 * BLK + j], a1 = Tbase[p1 * BLK + j];
        for (int k = 0; k < j; ++k) {
          const float gkj = 2.f * ga[k * BLK + j];
          a0 += tl0[k] * gkj;
          a1 += tl1[k] * gkj;
        }
        float tv0 = -ETA * a0, tv1 = -ETA * a1;
        if (first && j == 0) { tv0 = t0[p0]; tv1 = t0[p1]; }
        tl0[j] = tv0; tl1[j] = tv1;
        Tout[p0 * BLK + j] = tv0; Tout[p1 * BLK + j] = tv1;

        float tt = tv0 * tv0 + tv1 * tv1;
        for (int o = 16; o > 0; o >>= 1) tt += __shfl_xor(tt, o, 32);
        if (lane == 0) red[wave] = tt;
        __syncthreads();
        if (tid == 0) {
          float tsum = 0.f;
          for (int w = 0; w < 8; ++w) tsum += red[w];
          float sc;
          if (first && j == 0) {
            sc = s0[0];
          } else {
            float u = ub[j];
            for (int k = 0; k < j; ++k)
              u += sg[k] * (ga[k * BLK + j] + xl[k] * xl[j]);
            sc = -ETA * u + tsum;
          }
          const float sgn = 2.f * (sc - b[s + j]);
          sg[j] = sgn; signs[j] = sgn; preds[s + j] = sc;
        }
        __syncthreads();
      }
    }
    grid_sync(cnt, gen);

    // ---- phase C: g_W2 += 2*T @ A_blk^T (2048 tiles over 384 waves) ----
    for (int t = gw; t < 2048; t += 8 * NWG) {
      const int rt = t >> 6, ct = t & 63;
      const int nglob = ct * 16 + mrow;          // g_W2 column (feature index)
      const float msk = (nglob < DA) ? 1.f : 0.f;  // keep padded col 1023 at 0
      v8f c;
      for (int rr = 0; rr < 8; ++rr)
        c[rr] = gW2[(size_t)(rt * 16 + rr + 8 * half) * GW2_LD + ct * 16 + mrow];
      const float* trow = Tout + (rt * 16 + mrow) * BLK + 2 * half;
      const float* bsh  = Ash + nglob * BLK + 2 * half;
      for (int k = 0; k < BLK; k += 4) {
        v2f a  = {2.f * trow[k], 2.f * trow[k + 1]};
        v2f bb = {msk * bsh[k], msk * bsh[k + 1]};
        c = wmma_f32_4(a, bb, c);
      }
      for (int rr = 0; rr < 8; ++rr)
        gW2[(size_t)(rt * 16 + rr + 8 * half) * GW2_LD + ct * 16 + mrow] = c[rr];
    }
    if (g == NWG - 1) {                          // g_w1 += X_blk @ signs
      for (int q = 0; q < 4; ++q) {
        const int r = tid + 256 * q;
        float acc = gw1[r];
        for (int j = 0; j < BLK; ++j) acc += signs[j] * Ash[r * BLK + j];
        gw1[r] = acc;
      }
    }
    grid_sync(cnt, gen);
  }
}

// ---------------------------------------------------------------------------
extern "C" void kernel_launch(void* const* d_in, const int* in_sizes, int n_in,
                              void* d_out, int out_size, void* d_ws, size_t ws_size,
                              hipStream_t stream) {
  const float* At  = (const float*)d_in[0];   // (1024, 20000) row-major
  const float* b   = (const float*)d_in[1];   // (20000,)
  const float* w1i = (const float*)d_in[2];   // (1024,)
  const float* W2i = (const float*)d_in[3];   // (512, 1023) row-major
  float* out = (float*)d_out;                 // preds(20000) ++ b(20000)

  float* ws    = (float*)d_ws;
  float* gW2   = ws;                    // 512*1024 (col 1023 always 0)
  float* gw1   = gW2 + M2 * GW2_LD;     // 1024
  float* Tbase = gw1 + D;               // 512*32
  float* Tout  = Tbase + M2 * BLK;      // 512*32
  float* Ga    = Tout + M2 * BLK;       // 32*32
  float* ubase = Ga + BLK * BLK;        // 32
  float* xlast = ubase + BLK;           // 32
  float* signs = xlast + BLK;           // 32
  float* t0    = signs + BLK;           // 512
  float* s0    = t0 + M2;               // 1
  int*   cnt   = (int*)(s0 + 1);        // barrier words
  int*   gen   = cnt + 1;

  k_init<<<512, 256, 0, stream>>>(gW2, gw1, cnt, gen);
  k_step0<<<1, 512, 0, stream>>>(At, w1i, W2i, t0, s0);
  k_bcopy<<<(NSTEPS + 255) / 256, 256, 0, stream>>>(b, out);
  k_ftrl_main<<<NWG, 256, 0, stream>>>(At, b, gW2, gw1, Tbase, Ga, ubase, xlast,
                                       Tout, signs, t0, s0, out, cnt, gen);
}